// Router_5033701671233
// MI455X (gfx1250) — compile-verified
//
#include <hip/hip_runtime.h>
#include <math.h>

typedef __attribute__((ext_vector_type(2))) float v2f;
typedef __attribute__((ext_vector_type(8))) float v8f;

#define DIM 2048
#define NEXP 64
#define TOK_PER_BLOCK 64
#define NTHREADS 128

// ---------------------------------------------------------------------------
// Kernel 1: zero the usage histogram in workspace
// ---------------------------------------------------------------------------
__global__ void router_zero_usage(float* __restrict__ usage) {
    usage[threadIdx.x] = 0.0f;
}

// ---------------------------------------------------------------------------
// Kernel 2: fused GEMM (fp32 WMMA) + top-2 + weight normalization + histogram
// ---------------------------------------------------------------------------
__global__ void __launch_bounds__(NTHREADS)
router_gemm_top2(const float* __restrict__ x,       // [tokens, DIM]
                 const float* __restrict__ gw,      // [NEXP, DIM]
                 const float* __restrict__ gb,      // [NEXP]
                 float* __restrict__ out_idx,       // [tokens, 2] (as float)
                 float* __restrict__ out_w,         // [tokens, 2]
                 float* __restrict__ usage,         // [NEXP]
                 int tokens) {
    __shared__ float    lds_logits[TOK_PER_BLOCK * NEXP];  // 16 KB
    __shared__ unsigned lds_hist[NEXP];

    const int tid  = threadIdx.x;
    const int wave = tid >> 5;          // 0..3 (wave32)
    const int lane = tid & 31;
    const int half = lane >> 4;         // 0 or 1
    const int l16  = lane & 15;

    if (tid < NEXP) lds_hist[tid] = 0u;

    // ---- WMMA fp32 GEMM: 16 tokens x 64 experts per wave --------------------
    // A (16x4 f32): lane = M, VGPR0 holds K = 0 (half 0) / 2 (half 1),
    //               VGPR1 holds K = 1 / 3.
    // B (4x16 f32): lane = N, same K split as A (B = gate_w^T, row-major gw).
    const int tok0 = blockIdx.x * TOK_PER_BLOCK + wave * 16;

    const float* xrow = x  + (size_t)(tok0 + l16) * DIM + (half * 2);
    const float* g0   = gw + (size_t)l16 * DIM + (half * 2);
    const float* g1   = g0 + (size_t)16 * DIM;
    const float* g2   = g0 + (size_t)32 * DIM;
    const float* g3   = g0 + (size_t)48 * DIM;

    v8f acc0 = {}, acc1 = {}, acc2 = {}, acc3 = {};

#pragma unroll 4
    for (int k = 0; k < DIM; k += 4) {
        v2f a  = *(const v2f*)(xrow + k);
        v2f b0 = *(const v2f*)(g0 + k);
        v2f b1 = *(const v2f*)(g1 + k);
        v2f b2 = *(const v2f*)(g2 + k);
        v2f b3 = *(const v2f*)(g3 + k);
        // 8 args: (neg_a, A, neg_b, B, c_mod, C, reuse_a, reuse_b)
        acc0 = __builtin_amdgcn_wmma_f32_16x16x4_f32(false, a, false, b0,
                                                     (short)0, acc0, false, false);
        acc1 = __builtin_amdgcn_wmma_f32_16x16x4_f32(false, a, false, b1,
                                                     (short)0, acc1, false, false);
        acc2 = __builtin_amdgcn_wmma_f32_16x16x4_f32(false, a, false, b2,
                                                     (short)0, acc2, false, false);
        acc3 = __builtin_amdgcn_wmma_f32_16x16x4_f32(false, a, false, b3,
                                                     (short)0, acc3, false, false);
    }

    // ---- spill logits to LDS ------------------------------------------------
    // C/D layout: VGPR v, lane half h -> row M = v + 8h, col N = lane&15.
    const int row_base = wave * 16 + 8 * half;
#pragma unroll
    for (int v = 0; v < 8; ++v) {
        lds_logits[(row_base + v) * NEXP +  0 + l16] = acc0[v];
        lds_logits[(row_base + v) * NEXP + 16 + l16] = acc1[v];
        lds_logits[(row_base + v) * NEXP + 32 + l16] = acc2[v];
        lds_logits[(row_base + v) * NEXP + 48 + l16] = acc3[v];
    }
    __syncthreads();

    // ---- per-token top-2 + normalized weights + histogram -------------------
    if (tid < TOK_PER_BLOCK) {
        const int token = blockIdx.x * TOK_PER_BLOCK + tid;
        const float* row = &lds_logits[tid * NEXP];

        float best = -3.402823466e38f; int bi = 0;
#pragma unroll
        for (int e = 0; e < NEXP; ++e) {
            float v = row[e] + gb[e];
            if (v > best) { best = v; bi = e; }     // strict >: first index wins (JAX tie rule)
        }
        float sec = -3.402823466e38f; int si = 0;
#pragma unroll
        for (int e = 0; e < NEXP; ++e) {
            if (e == bi) continue;
            float v = row[e] + gb[e];
            if (v > sec) { sec = v; si = e; }
        }
        // normalized top-2 softmax weights: w1 = 1/(1+e^(l2-l1)), w2 = 1-w1
        float r  = __expf(sec - best);
        float w1 = 1.0f / (1.0f + r);
        float w2 = r / (1.0f + r);

        if (token < tokens) {
            out_idx[token * 2 + 0] = (float)bi;
            out_idx[token * 2 + 1] = (float)si;
            out_w[token * 2 + 0]   = w1;
            out_w[token * 2 + 1]   = w2;
            atomicAdd(&lds_hist[bi], 1u);
            atomicAdd(&lds_hist[si], 1u);
        }
    }
    __syncthreads();

    if (tid < NEXP) {
        unsigned c = lds_hist[tid];
        if (c) atomicAdd(&usage[tid], (float)c);
    }
}

// ---------------------------------------------------------------------------
// Kernel 3: load-balance loss = std(usage, ddof=1) / (mean + 1e-10) * 0.01
// ---------------------------------------------------------------------------
__global__ void router_loss(const float* __restrict__ usage,
                            float* __restrict__ loss_out) {
    __shared__ float s[NEXP];
    int tid = threadIdx.x;
    s[tid] = usage[tid];
    __syncthreads();
    if (tid == 0) {
        float sum = 0.0f;
        for (int i = 0; i < NEXP; ++i) sum += s[i];
        float mean = sum / (float)NEXP;
        float var = 0.0f;
        for (int i = 0; i < NEXP; ++i) {
            float d = s[i] - mean;
            var += d * d;
        }
        var /= (float)(NEXP - 1);                  // Bessel (ddof=1)
        loss_out[0] = sqrtf(var) / (mean + 1e-10f) * 0.01f;
    }
}

// ---------------------------------------------------------------------------
extern "C" void kernel_launch(void* const* d_in, const int* in_sizes, int n_in,
                              void* d_out, int out_size, void* d_ws, size_t ws_size,
                              hipStream_t stream) {
    const float* x  = (const float*)d_in[0];   // [4,4096,2048] fp32
    const float* gw = (const float*)d_in[1];   // [64,2048] fp32
    const float* gb = (const float*)d_in[2];   // [64] fp32

    const int tokens = in_sizes[0] / DIM;      // 16384

    float* out      = (float*)d_out;
    float* out_idx  = out;                                 // tokens*2 indices (as float)
    float* out_w    = out + (size_t)tokens * 2;            // tokens*2 weights
    float* loss     = out + (size_t)tokens * 4;            // 1 scalar
    float* usage    = (float*)d_ws;                        // 64 floats of scratch

    router_zero_usage<<<1, NEXP, 0, stream>>>(usage);

    const int blocks = (tokens + TOK_PER_BLOCK - 1) / TOK_PER_BLOCK;
    router_gemm_top2<<<blocks, NTHREADS, 0, stream>>>(x, gw, gb, out_idx, out_w,
                                                      usage, tokens);

    router_loss<<<1, NEXP, 0, stream>>>(usage, loss);
}